// HungarianMatcher_78365973282877
// MI455X (gfx1250) — compile-verified
//
#include <hip/hip_runtime.h>
#include <hip/hip_bf16.h>
#include <math.h>

#define BN       128   // preds per block (rows)
#define BM       512   // targets per block (cols)
#define THREADS  256   // 8 wave32
#define NCLS_MAX 64

// CDNA5 async global->LDS copy, 16 bytes per lane, GV addressing mode.
// vdst = per-lane LDS byte address, vaddr = per-lane 64-bit global address.
__device__ __forceinline__ void async_copy_b128(unsigned lds_off, const void* gptr) {
    unsigned long long ga = (unsigned long long)(size_t)gptr;
    asm volatile("global_load_async_to_lds_b128 %0, %1, off"
                 :: "v"(lds_off), "v"(ga) : "memory");
}

__device__ __forceinline__ void wait_asynccnt0() {
    asm volatile("s_wait_asynccnt 0" ::: "memory");
}

__global__ __launch_bounds__(THREADS)
void matcher_cost_kernel(const float* __restrict__ cls_boxes,   // [NPRED, NCLS]
                         const int*   __restrict__ cls_targets, // [NTGT] (1-based)
                         const float* __restrict__ pred_boxes,  // [NPRED, 4]
                         const float* __restrict__ tgt_boxes,   // [NTGT, 4]
                         float*       __restrict__ out,         // [NPRED, NTGT]
                         int NPRED, int NTGT, int NCLS) {
    __shared__ __align__(16) float s_pred[BN * 4];
    __shared__ __align__(16) float s_tgt [BM * 4];
    __shared__ float s_sig[BN * NCLS_MAX]; // (1 - sigmoid) table per pred row
    __shared__ int   s_cls[BM];            // 0-based target class

    const int tid  = threadIdx.x;
    const int row0 = blockIdx.y * BN;
    const int col0 = blockIdx.x * BM;

    // ---- Stage 1: kick off async DMA of the box tiles into LDS ----------
    if (tid < BN) {
        int n = row0 + tid; if (n >= NPRED) n = NPRED - 1;
        unsigned loff = (unsigned)(size_t)(&s_pred[tid * 4]);
        async_copy_b128(loff, pred_boxes + (size_t)n * 4);
    }
    #pragma unroll
    for (int m = tid; m < BM; m += THREADS) {
        int mg = col0 + m; if (mg >= NTGT) mg = NTGT - 1;
        unsigned loff = (unsigned)(size_t)(&s_tgt[m * 4]);
        async_copy_b128(loff, tgt_boxes + (size_t)mg * 4);
    }

    // ---- Stage 2 (overlapped with DMA): class ids + sigmoid table -------
    #pragma unroll
    for (int m = tid; m < BM; m += THREADS) {
        int mg = col0 + m; if (mg >= NTGT) mg = NTGT - 1;
        s_cls[m] = cls_targets[mg] - 1;   // labels are 1..NCLS
    }
    const int tabN = BN * NCLS;
    for (int idx = tid; idx < tabN; idx += THREADS) {
        int nl = idx / NCLS;
        int c  = idx - nl * NCLS;
        int n  = row0 + nl; if (n >= NPRED) n = NPRED - 1;
        float x = cls_boxes[(size_t)n * NCLS + c];
        // cost_cls = 1 - sigmoid(x) = 1 / (1 + exp(x))
        s_sig[idx] = __fdividef(1.0f, 1.0f + __expf(x));
    }

    wait_asynccnt0();      // ASYNCcnt == 0: this wave's DMA landed in LDS
    __syncthreads();       // make all tiles visible workgroup-wide

    // ---- Stage 3: compute 128 x 512 tile, 4 targets per thread ----------
    const int col4 = tid & (BM / 4 - 1);           // 0..127 -> 4 targets each
    float tx1[4], ty1[4], tx2[4], ty2[4], tarea[4];
    int   tcls[4];
    #pragma unroll
    for (int j = 0; j < 4; ++j) {
        int m = col4 * 4 + j;
        float x1 = s_tgt[m * 4 + 0], y1 = s_tgt[m * 4 + 1];
        float x2 = s_tgt[m * 4 + 2], y2 = s_tgt[m * 4 + 3];
        tx1[j] = x1; ty1[j] = y1; tx2[j] = x2; ty2[j] = y2;
        tarea[j] = (x2 - x1) * (y2 - y1);
        tcls[j]  = s_cls[m];
    }

    const size_t ocol = (size_t)col0 + (size_t)col4 * 4;
    for (int r = tid >> 7; r < BN; r += (THREADS * 4) / BM) {  // step 2
        const int n = row0 + r;
        float px1 = s_pred[r * 4 + 0], py1 = s_pred[r * 4 + 1];
        float px2 = s_pred[r * 4 + 2], py2 = s_pred[r * 4 + 3];
        float parea = (px2 - px1) * (py2 - py1);
        const float* sigrow = &s_sig[r * NCLS];

        float res[4];
        #pragma unroll
        for (int j = 0; j < 4; ++j) {
            float l1 = fabsf(px1 - tx1[j]) + fabsf(py1 - ty1[j]) +
                       fabsf(px2 - tx2[j]) + fabsf(py2 - ty2[j]);
            float iw = fminf(px2, tx2[j]) - fmaxf(px1, tx1[j]);
            float ih = fminf(py2, ty2[j]) - fmaxf(py1, ty1[j]);
            iw = fmaxf(iw, 0.0f);
            ih = fmaxf(ih, 0.0f);
            float ov = iw * ih;
            float un = fmaxf(parea + tarea[j] - ov, 1e-6f);
            float iou = __fdividef(ov, un);
            res[j] = sigrow[tcls[j]] + l1 - iou;
        }
        if (n < NPRED && ocol + 4 <= (size_t)NTGT) {
            float4 o = make_float4(res[0], res[1], res[2], res[3]);
            *reinterpret_cast<float4*>(out + (size_t)n * NTGT + ocol) = o;
        }
    }
}

extern "C" void kernel_launch(void* const* d_in, const int* in_sizes, int n_in,
                              void* d_out, int out_size, void* d_ws, size_t ws_size,
                              hipStream_t stream) {
    const float* cls_boxes   = (const float*)d_in[0];
    const int*   cls_targets = (const int*)  d_in[1];
    const float* pred_boxes  = (const float*)d_in[2];
    const float* tgt_boxes   = (const float*)d_in[3];
    float*       out         = (float*)d_out;

    const int NTGT  = in_sizes[1];
    const int NPRED = in_sizes[2] / 4;
    const int NCLS  = in_sizes[0] / NPRED;   // 40

    dim3 grid((NTGT + BM - 1) / BM, (NPRED + BN - 1) / BN);  // (4, 128)
    matcher_cost_kernel<<<grid, THREADS, 0, stream>>>(
        cls_boxes, cls_targets, pred_boxes, tgt_boxes, out, NPRED, NTGT, NCLS);
}